// AggFeatureModel_51135880626856
// MI455X (gfx1250) — compile-verified
//
#include <hip/hip_runtime.h>
#include <hip/hip_bf16.h>
#include <math.h>

#define BDIM   256
#define TT     2048
#define CHUNK  1024     // elements staged per buffer (256 lanes * 4)
#define NMCC   100
#define NTR    50
#define EPSF   1e-9f

// Async global -> LDS copy of 16 bytes per lane (gfx1250 GLOBAL_LOAD_ASYNC_TO_LDS_B128).
// VDST = VGPR holding workgroup-relative LDS byte offset, VADDR = 64-bit global address.
// Tracked with ASYNCcnt (per-wave, in-order completion for loads).
__device__ __forceinline__ void async_copy16(void* lds_ptr, const void* gptr) {
    uint32_t lds_off = (uint32_t)(uintptr_t)lds_ptr;   // low 32 bits of LDS-aperture flat addr = LDS offset
    uint64_t gaddr   = (uint64_t)(uintptr_t)gptr;
    asm volatile("global_load_async_to_lds_b128 %0, %1, off"
                 :: "v"(lds_off), "v"(gaddr)
                 : "memory");
}

template <int N>
__device__ __forceinline__ void wait_async() {
    asm volatile("s_wait_asynccnt %0" :: "i"(N) : "memory");
}

__device__ __forceinline__ void lds_fadd(float* p, float v) {
    (void)__hip_atomic_fetch_add(p, v, __ATOMIC_RELAXED, __HIP_MEMORY_SCOPE_WORKGROUP);
}

// Accumulator LDS layout (floats):
// [0,100)   mcc count | [100,200) mcc sum | [200,300) mcc sum-of-squares
// [300,350) tr  count | [350,400) tr  sum | [400,450) tr  sum-of-squares
// [450] row sum  [451] row sum-of-squares
__global__ __launch_bounds__(BDIM)
void agg_feature_kernel(const float* __restrict__ amount,
                        const int*   __restrict__ mcc,
                        const int*   __restrict__ tr,
                        const int*   __restrict__ seq_lens,
                        float*       __restrict__ out)
{
    __shared__ __align__(16) float s_amt[2][CHUNK];
    __shared__ __align__(16) int   s_mcc[2][CHUNK];
    __shared__ __align__(16) int   s_tr [2][CHUNK];
    __shared__ float acc[452];

    const int b   = blockIdx.x;
    const int tid = threadIdx.x;
    const long rowBase = (long)b * TT;
    const int lane4 = tid * 4;

    // Stage both chunks of all three inputs via the async data mover (6 async ops per wave,
    // issued in chunk order so ASYNCcnt<=3 means chunk 0 has fully landed for this wave).
#pragma unroll
    for (int buf = 0; buf < 2; ++buf) {
        const long idx = rowBase + buf * CHUNK + lane4;
        async_copy16(&s_amt[buf][lane4], amount + idx);
        async_copy16(&s_mcc[buf][lane4], mcc    + idx);
        async_copy16(&s_tr [buf][lane4], tr     + idx);
    }

    // Zero accumulators while the DMA is in flight.
    for (int i = tid; i < 452; i += BDIM) acc[i] = 0.0f;
    __syncthreads();

    float lsum = 0.0f, lss = 0.0f;
#pragma unroll
    for (int buf = 0; buf < 2; ++buf) {
        if (buf == 0) wait_async<3>();   // chunk 0 landed; chunk 1 still streaming
        else          wait_async<0>();
#pragma unroll
        for (int k = 0; k < 4; ++k) {
            const int j = lane4 + k;     // each lane consumes exactly the bytes it staged
            const float a  = s_amt[buf][j];
            const int   m  = s_mcc[buf][j];
            const int   t  = s_tr [buf][j];
            const float a2 = a * a;
            lsum += a;
            lss  += a2;
            lds_fadd(&acc[m],        1.0f);
            lds_fadd(&acc[100 + m],  a);
            lds_fadd(&acc[200 + m],  a2);
            lds_fadd(&acc[300 + t],  1.0f);
            lds_fadd(&acc[350 + t],  a);
            lds_fadd(&acc[400 + t],  a2);
        }
    }
    lds_fadd(&acc[450], lsum);
    lds_fadd(&acc[451], lss);
    __syncthreads();

    float* o = out + (long)b * 456;

    if (tid == 0) {
        const float sl = (float)seq_lens[b];
        const float S  = acc[450];
        const float SS = acc[451];
        o[0] = sl;
        o[1] = S;
        o[2] = S / (sl + EPSF);
        float a = SS - S * S / (sl + EPSF);
        a = fmaxf(a, 0.0f);
        o[3] = sqrtf(a / (fmaxf(sl - 1.0f, 0.0f) + EPSF));
    }
    if (tid < NMCC) {
        const int c = tid;
        const float cnt = acc[c];
        const float s   = acc[100 + c];
        const float ss  = acc[200 + c];
        const float ecnt = (c > 0) ? cnt : 0.0f;          // mask bin 0 (padding)
        o[4 + c]   = ecnt;
        o[104 + c] = s / (ecnt + EPSF);
        float a = ss - s * s / (ecnt + EPSF);
        a = fmaxf(a, 0.0f);
        o[204 + c] = sqrtf(a / (fmaxf(ecnt - 1.0f, 0.0f) + EPSF));
    } else if (tid < NMCC + NTR) {
        const int c = tid - NMCC;
        const float cnt = acc[300 + c];
        const float s   = acc[350 + c];
        const float ss  = acc[400 + c];
        const float ecnt = (c > 0) ? cnt : 0.0f;
        o[304 + c] = ecnt;
        o[354 + c] = s / (ecnt + EPSF);
        float a = ss - s * s / (ecnt + EPSF);
        a = fmaxf(a, 0.0f);
        o[404 + c] = sqrtf(a / (fmaxf(ecnt - 1.0f, 0.0f) + EPSF));
    } else if (tid == NMCC + NTR) {
        float d = 0.0f;
        for (int c = 1; c < NMCC; ++c) d += (acc[c] > 0.0f) ? 1.0f : 0.0f;
        o[454] = d;
    } else if (tid == NMCC + NTR + 1) {
        float d = 0.0f;
        for (int c = 1; c < NTR; ++c) d += (acc[300 + c] > 0.0f) ? 1.0f : 0.0f;
        o[455] = d;
    }
}

extern "C" void kernel_launch(void* const* d_in, const int* in_sizes, int n_in,
                              void* d_out, int out_size, void* d_ws, size_t ws_size,
                              hipStream_t stream) {
    const float* amount   = (const float*)d_in[0];
    const int*   mcc      = (const int*)  d_in[1];
    const int*   tr_type  = (const int*)  d_in[2];
    const int*   seq_lens = (const int*)  d_in[3];
    float*       out      = (float*)d_out;

    const int B = in_sizes[3];   // 4096 rows, one workgroup (8 wave32) per row
    agg_feature_kernel<<<B, BDIM, 0, stream>>>(amount, mcc, tr_type, seq_lens, out);
}